// TextDecoder_80762565034104
// MI455X (gfx1250) — compile-verified
//
#include <hip/hip_runtime.h>
#include <hip/hip_bf16.h>
#include <math.h>

typedef __attribute__((ext_vector_type(2))) float v2f;
typedef __attribute__((ext_vector_type(8))) float v8f;

#define B_ 64
#define H_ 300
#define V_ 32000
#define T_ 48
#define G3 900          // 3*H
#define KSTEPS 75       // 300 / 4
#define NT_PER_BLK 8    // one 16-col N-tile per wave, 8 waves/block
#define BLK_COLS 128    // 8 * 16
#define CSTRIDE 132     // padded LDS row stride (avoid bank conflicts)

// ---------------------------------------------------------------- init
__global__ __launch_bounds__(256) void init_state(const int* __restrict__ tokens_in,
                                                  const float* __restrict__ h0,
                                                  float* __restrict__ h,
                                                  int* __restrict__ tok) {
  int i = blockIdx.x * 256 + threadIdx.x;
  if (i < B_ * H_) h[i] = h0[i];
  if (i < B_) tok[i] = tokens_in[i];
}

// ---------------------------------------------------------------- embed
__global__ __launch_bounds__(256) void embed_lookup(const int* __restrict__ tok,
                                                    const float* __restrict__ embed,
                                                    float* __restrict__ x) {
  int i = blockIdx.x * 256 + threadIdx.x;
  if (i < B_ * H_) {
    int b = i / H_, k = i % H_;
    x[i] = embed[(size_t)tok[b] * H_ + k];
  }
}

// ---------------------------------------------------------------- WMMA GEMM
// C[64, N] = A[64, 300] @ W[N, 300]^T + bias[N]
// Optional softmax/argmax partials per (row, block) when `partials` != null.
__global__ __launch_bounds__(256) void gemm_wmma(const float* __restrict__ A,
                                                 const float* __restrict__ W,
                                                 const float* __restrict__ bias,
                                                 float* __restrict__ C, int N,
                                                 float* __restrict__ partials) {
  __shared__ float csh[B_ * CSTRIDE];

  const int tid  = threadIdx.x;
  const int wave = tid >> 5;
  const int lane = tid & 31;
  const int l16  = lane & 15;
  const int lh   = lane >> 4;  // lane half selects K-pair (ISA fp32 A/B layout)

  const int ntile    = blockIdx.x * NT_PER_BLK + wave;
  const int n_global = ntile * 16 + l16;
  const int nc       = (n_global < N) ? n_global : (N - 1);  // clamp edge tiles
  const bool do_part = (partials != nullptr);

  v8f acc[4] = {};

  const float* Wrow = W + (size_t)nc * H_;
  for (int kk = 0; kk < KSTEPS; ++kk) {
    const int kb = kk * 4 + lh * 2;   // this lane's K-pair base
    v2f bfrag = *(const v2f*)(Wrow + kb);        // B[k..k+1, n] contiguous
#pragma unroll
    for (int mt = 0; mt < 4; ++mt) {
      const int m = mt * 16 + l16;
      v2f afrag = *(const v2f*)(A + m * H_ + kb);  // A[m, k..k+1]
      // (neg_a, A, neg_b, B, c_mod, C, reuse_a, reuse_b)
      acc[mt] = __builtin_amdgcn_wmma_f32_16x16x4_f32(
          false, afrag, false, bfrag, (short)0, acc[mt], false, false);
    }
  }

  const float bv = (n_global < N) ? bias[n_global] : 0.0f;
#pragma unroll
  for (int mt = 0; mt < 4; ++mt) {
#pragma unroll
    for (int i = 0; i < 8; ++i) {
      const int m = mt * 16 + lh * 8 + i;   // C/D VGPR layout: lanes16-31 -> M+8
      const float val = acc[mt][i] + bv;
      if (n_global < N) C[(size_t)m * N + n_global] = val;
      if (do_part) csh[m * CSTRIDE + wave * 16 + l16] = val;
    }
  }

  if (do_part) {
    __syncthreads();
    if (tid < B_) {
      const int row = tid;
      float M = -__builtin_inff(), S = 0.0f;
      int bi = 0;
      for (int j = 0; j < BLK_COLS; ++j) {
        const float v = csh[row * CSTRIDE + j];
        if (v > M) { S = S * __expf(M - v) + 1.0f; M = v; bi = j; }
        else       { S += __expf(v - M); }
      }
      float* p = partials + ((size_t)row * gridDim.x + blockIdx.x) * 4;
      p[0] = M;
      p[1] = S;
      p[2] = (float)(blockIdx.x * BLK_COLS + bi);
    }
  }
}

// ---------------------------------------------------------------- GRU gates
__global__ __launch_bounds__(256) void gru_gates(const float* __restrict__ gi,
                                                 const float* __restrict__ gh,
                                                 float* __restrict__ h) {
  int i = blockIdx.x * 256 + threadIdx.x;
  if (i >= B_ * H_) return;
  int b = i / H_, j = i % H_;
  const float* gib = gi + (size_t)b * G3;
  const float* ghb = gh + (size_t)b * G3;
  float r = 1.0f / (1.0f + __expf(-(gib[j] + ghb[j])));
  float z = 1.0f / (1.0f + __expf(-(gib[H_ + j] + ghb[H_ + j])));
  float n = tanhf(gib[2 * H_ + j] + r * ghb[2 * H_ + j]);
  h[i] = (1.0f - z) * n + z * h[i];
}

// ---------------------------------------------------------------- softmax reduce
__global__ __launch_bounds__(64) void softmax_reduce(const float* __restrict__ partials,
                                                     int nblk,
                                                     float* __restrict__ logZ,
                                                     int* __restrict__ tok,
                                                     float* __restrict__ ids_out,
                                                     int t) {
  const int row = threadIdx.x;  // 64 threads = 64 rows
  float M = -__builtin_inff(), S = 0.0f;
  float bv = -__builtin_inff(), bi = 0.0f;
  for (int b = 0; b < nblk; ++b) {
    const float* p = partials + ((size_t)row * nblk + b) * 4;
    const float pm = p[0], ps = p[1], pi = p[2];
    if (pm > M) { S = S * __expf(M - pm) + ps; M = pm; }
    else        { S += ps * __expf(pm - M); }
    if (pm > bv) { bv = pm; bi = pi; }  // first-occurrence argmax
  }
  logZ[row] = M + __logf(S);
  const int pred = (int)bi;
  tok[row] = pred;
  ids_out[(size_t)row * T_ + t] = (float)pred;
}

// ---------------------------------------------------------------- log-prob writeout
__global__ __launch_bounds__(256) void logp_writeout(const float* __restrict__ logits,
                                                     const float* __restrict__ logZ,
                                                     float* __restrict__ out, int t) {
  size_t i = (size_t)blockIdx.x * 256 + threadIdx.x;
  if (i >= (size_t)B_ * V_) return;
  const int b = (int)(i / V_);
  const int v = (int)(i % V_);
  out[(size_t)b * T_ * V_ + (size_t)t * V_ + v] = logits[i] - logZ[b];
}

// ---------------------------------------------------------------- launch
extern "C" void kernel_launch(void* const* d_in, const int* in_sizes, int n_in,
                              void* d_out, int out_size, void* d_ws, size_t ws_size,
                              hipStream_t stream) {
  const int*   input_tokens = (const int*)d_in[0];
  const float* h0           = (const float*)d_in[1];
  // d_in[2] = max_len (scalar; compile-time T_=48)
  const float* embed        = (const float*)d_in[3];
  const float* w_ih         = (const float*)d_in[4];
  const float* w_hh         = (const float*)d_in[5];
  const float* b_ih         = (const float*)d_in[6];
  const float* b_hh         = (const float*)d_in[7];
  const float* w_out        = (const float*)d_in[8];
  const float* b_out        = (const float*)d_in[9];

  const int NBLK_V = V_ / BLK_COLS;                 // 250
  const int NBLK_G = (G3 + BLK_COLS - 1) / BLK_COLS; // 8 (covers 57 tiles w/ clamp)

  float* ws       = (float*)d_ws;
  float* h        = ws;                              // [64*300]
  float* x        = h + B_ * H_;                     // [64*300]
  float* gi       = x + B_ * H_;                     // [64*900]
  float* gh       = gi + (size_t)B_ * G3;            // [64*900]
  float* logits   = gh + (size_t)B_ * G3;            // [64*32000]
  float* partials = logits + (size_t)B_ * V_;        // [64*250*4]
  float* logZ     = partials + (size_t)B_ * NBLK_V * 4; // [64]
  int*   tok      = (int*)(logZ + B_);               // [64]

  float* out_logp = (float*)d_out;                   // [B, T, V]
  float* out_ids  = out_logp + (size_t)B_ * T_ * V_; // [B, T] (as float)

  init_state<<<76, 256, 0, stream>>>(input_tokens, h0, h, tok);

  for (int t = 0; t < T_; ++t) {
    embed_lookup<<<75, 256, 0, stream>>>(tok, embed, x);
    gemm_wmma<<<NBLK_G, 256, 0, stream>>>(x, w_ih, b_ih, gi, G3, nullptr);
    gemm_wmma<<<NBLK_G, 256, 0, stream>>>(h, w_hh, b_hh, gh, G3, nullptr);
    gru_gates<<<75, 256, 0, stream>>>(gi, gh, h);
    gemm_wmma<<<NBLK_V, 256, 0, stream>>>(h, w_out, b_out, logits, V_, partials);
    softmax_reduce<<<1, 64, 0, stream>>>(partials, NBLK_V, logZ, tok, out_ids, t);
    logp_writeout<<<8000, 256, 0, stream>>>(logits, logZ, out_logp, t);
  }
}